// SimpleDynamicMoEModel_64725157151314
// MI455X (gfx1250) — compile-verified
//
#include <hip/hip_runtime.h>
#include <hip/hip_bf16.h>

// ---------------------------------------------------------------------------
// MoE block for MI455X (gfx1250): fp32 WMMA (v_wmma_f32_16x16x4_f32) GEMMs,
// async global->LDS double-buffered tiles (ASYNCcnt), token dispatch by
// expert, LDS-tiled 64x128 output tiles, wave32.
// ---------------------------------------------------------------------------

typedef float v2f __attribute__((ext_vector_type(2)));
typedef float v8f __attribute__((ext_vector_type(8)));

#define T_TOK   16384     // B*S = 8*2048
#define D       512
#define NEXP    8
#define BM      64
#define BN      128
#define BK      32
#define AST     40        // LDS row stride (floats), A tile [M][K], 16B-aligned rows
#define BSTNK   40        // LDS row stride (floats), B tile [N][K] (linear1)
#define BSTKN   144       // LDS row stride (floats), B tile [K][N] (moe); 2*144%64==32

// Async global->LDS copy, 16B per lane per issue. vdst = LDS byte address,
// vaddr = 64-bit global address (GV mode).
#define ASYNC_LOAD_B128(ldsOff, gptr)                                        \
  asm volatile("global_load_async_to_lds_b128 %0, %1, off"                   \
               :: "v"(ldsOff), "v"((unsigned long long)(uintptr_t)(gptr))    \
               : "memory")
#define ASYNC_WAIT() asm volatile("s_wait_asynccnt 0x0" ::: "memory")

__device__ __forceinline__ unsigned int lds_off32(const void* p) {
  // Generic pointers to LDS carry the wave-relative offset in the low 32 bits
  // (aperture in the high bits); async-load VDST wants the 32-bit LDS address.
  return (unsigned int)(uintptr_t)p;
}

// ---------------- GEMM 1: C = X * W^T + bias   (h = x @ W1.T + b1) ----------
__global__ __launch_bounds__(256) void linear1_kernel(
    const float* __restrict__ X, const float* __restrict__ W,
    const float* __restrict__ bias, float* __restrict__ C) {
  __shared__ __align__(16) float As[2][BM * AST];
  __shared__ __align__(16) float Bs[2][BN * BSTNK];
  const int tid  = threadIdx.x;
  const int lane = tid & 31;
  const int wave = tid >> 5;
  const int wm   = wave & 3;     // 4 waves along M
  const int wn   = wave >> 2;    // 2 waves along N (64 cols each)
  const int l15  = lane & 15;
  const int koff = (lane < 16) ? 0 : 2;   // K half-split across lane groups
  const int mBase = blockIdx.y * BM;
  const int nBase = blockIdx.x * BN;

  // This thread's fixed copy coordinates (2 x A, 4 x B float4 slots)
  int rA[2], cA[2], rB[4], cB[4];
#pragma unroll
  for (int i = 0; i < 2; ++i) { int idx = tid + i * 256; rA[i] = idx >> 3; cA[i] = (idx & 7) * 4; }
#pragma unroll
  for (int i = 0; i < 4; ++i) { int idx = tid + i * 256; rB[i] = idx >> 3; cB[i] = (idx & 7) * 4; }

  unsigned int aOff[2][2], bOff[2][4];
#pragma unroll
  for (int buf = 0; buf < 2; ++buf) {
#pragma unroll
    for (int i = 0; i < 2; ++i) aOff[buf][i] = lds_off32(&As[buf][rA[i] * AST + cA[i]]);
#pragma unroll
    for (int i = 0; i < 4; ++i) bOff[buf][i] = lds_off32(&Bs[buf][rB[i] * BSTNK + cB[i]]);
  }

  v8f acc[4] = {v8f{}, v8f{}, v8f{}, v8f{}};

  // Prologue: stage tile k0=0 into buffer 0
#pragma unroll
  for (int i = 0; i < 2; ++i)
    ASYNC_LOAD_B128(aOff[0][i], X + (size_t)(mBase + rA[i]) * D + cA[i]);
#pragma unroll
  for (int i = 0; i < 4; ++i)
    ASYNC_LOAD_B128(bOff[0][i], W + (size_t)(nBase + rB[i]) * D + cB[i]);
  ASYNC_WAIT();
  __syncthreads();

  for (int k0 = 0; k0 < D; k0 += BK) {
    const int cur = (k0 >> 5) & 1;
    const bool more = (k0 + BK) < D;
    if (more) {  // stage next tile into the other buffer (overlaps compute)
#pragma unroll
      for (int i = 0; i < 2; ++i)
        ASYNC_LOAD_B128(aOff[1 - cur][i], X + (size_t)(mBase + rA[i]) * D + k0 + BK + cA[i]);
#pragma unroll
      for (int i = 0; i < 4; ++i)
        ASYNC_LOAD_B128(bOff[1 - cur][i], W + (size_t)(nBase + rB[i]) * D + k0 + BK + cB[i]);
    }
    const float* Ab = &As[cur][(wm * 16 + l15) * AST + koff];
    const float* Bb = &Bs[cur][(wn * 64 + l15) * BSTNK + koff];
#pragma unroll
    for (int kk = 0; kk < BK; kk += 4) {
      v2f a = *(const v2f*)(Ab + kk);
#pragma unroll
      for (int j = 0; j < 4; ++j) {
        v2f b = *(const v2f*)(Bb + j * 16 * BSTNK + kk);
        acc[j] = __builtin_amdgcn_wmma_f32_16x16x4_f32(
            false, a, false, b, (short)0, acc[j], false, false);
      }
    }
    if (more) ASYNC_WAIT();
    __syncthreads();
  }

  const int rsel = (lane < 16) ? 0 : 8;
  float bv[4];
#pragma unroll
  for (int j = 0; j < 4; ++j) bv[j] = bias[nBase + wn * 64 + j * 16 + l15];
#pragma unroll
  for (int i = 0; i < 8; ++i) {
    size_t row = (size_t)(mBase + wm * 16 + i + rsel) * D + nBase + wn * 64 + l15;
#pragma unroll
    for (int j = 0; j < 4; ++j) C[row + j * 16] = acc[j][i] + bv[j];
  }
}

// ---------------- Gate: logits = H @ gw ; top-1 softmax ---------------------
__global__ __launch_bounds__(256) void gate_kernel(
    const float* __restrict__ H, const float* __restrict__ gw,
    int* __restrict__ idxOut, float* __restrict__ pOut) {
  __shared__ float gws[D * NEXP];  // 16 KB
  const int tid = threadIdx.x;
  for (int i = tid; i < D * NEXP; i += 256) gws[i] = gw[i];
  __syncthreads();
  const int lane = tid & 31, wave = tid >> 5;
  const int t = blockIdx.x * 8 + wave;
  const float* hrow = H + (size_t)t * D;
  float acc[NEXP] = {0.f, 0.f, 0.f, 0.f, 0.f, 0.f, 0.f, 0.f};
  for (int i = 0; i < D / 32; ++i) {
    int d = lane + i * 32;
    float hv = hrow[d];
#pragma unroll
    for (int e = 0; e < NEXP; ++e) acc[e] += hv * gws[d * NEXP + e];
  }
#pragma unroll
  for (int e = 0; e < NEXP; ++e)
    for (int off = 16; off > 0; off >>= 1)
      acc[e] += __shfl_xor(acc[e], off, 32);
  if (lane == 0) {
    float m = acc[0]; int bi = 0;
#pragma unroll
    for (int e = 1; e < NEXP; ++e)
      if (acc[e] > m) { m = acc[e]; bi = e; }
    float s = 0.f;
#pragma unroll
    for (int e = 0; e < NEXP; ++e) s += expf(acc[e] - m);
    idxOut[t] = bi;
    pOut[t]   = 1.0f / s;   // softmax prob of the argmax
  }
}

// ---------------- Dispatch: histogram / scan / scatter ----------------------
__global__ void histo_kernel(const int* __restrict__ idx, int* __restrict__ counts) {
  int t = blockIdx.x * 256 + threadIdx.x;
  atomicAdd(&counts[idx[t]], 1);
}
__global__ void scan_kernel(const int* __restrict__ counts,
                            int* __restrict__ bases, int* __restrict__ cursor) {
  if (threadIdx.x == 0 && blockIdx.x == 0) {
    int acc = 0;
    for (int e = 0; e < NEXP; ++e) { bases[e] = acc; cursor[e] = acc; acc += counts[e]; }
  }
}
__global__ void scatter_kernel(const int* __restrict__ idx,
                               int* __restrict__ cursor, int* __restrict__ perm) {
  int t = blockIdx.x * 256 + threadIdx.x;
  int e = idx[t];
  int pos = atomicAdd(&cursor[e], 1);
  perm[pos] = t;
}

// ---------------- Grouped expert GEMM: out[t] = p[t]*(in[t]@We[e] + be[e]) --
__global__ __launch_bounds__(256) void moe_kernel(
    const float* __restrict__ In, const float* __restrict__ We,
    const float* __restrict__ be, const int* __restrict__ perm,
    const int* __restrict__ bases, const int* __restrict__ counts,
    const float* __restrict__ topP, float* __restrict__ Out) {
  const int e = blockIdx.z;
  const int cnt = counts[e];
  const int mTile = blockIdx.y;
  if (mTile * BM >= cnt) return;

  __shared__ __align__(16) float As[2][BM * AST];
  __shared__ __align__(16) float Bs[2][BK * BSTKN];   // [K][N] layout
  __shared__ int rows[BM];

  const int tid  = threadIdx.x;
  const int lane = tid & 31;
  const int wave = tid >> 5;
  const int wm   = wave & 3;
  const int wn   = wave >> 2;
  const int l15  = lane & 15;
  const int koff = (lane < 16) ? 0 : 2;
  const int nBase = blockIdx.x * BN;

  if (tid < BM) {
    int p = mTile * BM + tid;
    rows[tid] = (p < cnt) ? perm[bases[e] + p] : -1;
  }
  __syncthreads();

  // This thread's fixed copy coordinates
  int rA[2], cA[2], kB[4], nB[4], trc[2];
#pragma unroll
  for (int i = 0; i < 2; ++i) {
    int idx = tid + i * 256;
    rA[i] = idx >> 3; cA[i] = (idx & 7) * 4;
    int tr = rows[rA[i]];
    trc[i] = (tr >= 0) ? tr : 0;  // clamp: padded rows load valid-but-unused data
  }
#pragma unroll
  for (int i = 0; i < 4; ++i) {
    int idx = tid + i * 256;
    kB[i] = idx >> 5;            // 0..31
    nB[i] = (idx & 31) * 4;      // 0..124
  }
  unsigned int aOff[2][2], bOff[2][4];
#pragma unroll
  for (int buf = 0; buf < 2; ++buf) {
#pragma unroll
    for (int i = 0; i < 2; ++i) aOff[buf][i] = lds_off32(&As[buf][rA[i] * AST + cA[i]]);
#pragma unroll
    for (int i = 0; i < 4; ++i) bOff[buf][i] = lds_off32(&Bs[buf][kB[i] * BSTKN + nB[i]]);
  }

  const float* Wexp = We + (size_t)e * D * D;
  v8f acc[4] = {v8f{}, v8f{}, v8f{}, v8f{}};

  // Prologue: stage tile k0=0 into buffer 0
#pragma unroll
  for (int i = 0; i < 2; ++i)
    ASYNC_LOAD_B128(aOff[0][i], In + (size_t)trc[i] * D + cA[i]);
#pragma unroll
  for (int i = 0; i < 4; ++i)
    ASYNC_LOAD_B128(bOff[0][i], Wexp + (size_t)kB[i] * D + nBase + nB[i]);
  ASYNC_WAIT();
  __syncthreads();

  for (int k0 = 0; k0 < D; k0 += BK) {
    const int cur = (k0 >> 5) & 1;
    const bool more = (k0 + BK) < D;
    if (more) {
#pragma unroll
      for (int i = 0; i < 2; ++i)
        ASYNC_LOAD_B128(aOff[1 - cur][i], In + (size_t)trc[i] * D + k0 + BK + cA[i]);
#pragma unroll
      for (int i = 0; i < 4; ++i)
        ASYNC_LOAD_B128(bOff[1 - cur][i],
                        Wexp + (size_t)(k0 + BK + kB[i]) * D + nBase + nB[i]);
    }
    const float* Ab = &As[cur][(wm * 16 + l15) * AST + koff];
    const float* Bb = &Bs[cur][koff * BSTKN + wn * 64 + l15];
#pragma unroll
    for (int kk = 0; kk < BK; kk += 4) {
      v2f a = *(const v2f*)(Ab + kk);
#pragma unroll
      for (int j = 0; j < 4; ++j) {
        v2f b;
        b.x = Bb[(kk + 0) * BSTKN + j * 16];
        b.y = Bb[(kk + 1) * BSTKN + j * 16];
        acc[j] = __builtin_amdgcn_wmma_f32_16x16x4_f32(
            false, a, false, b, (short)0, acc[j], false, false);
      }
    }
    if (more) ASYNC_WAIT();
    __syncthreads();
  }

  const int rsel = (lane < 16) ? 0 : 8;
  float bv[4];
#pragma unroll
  for (int j = 0; j < 4; ++j) bv[j] = be[e * D + nBase + wn * 64 + j * 16 + l15];
#pragma unroll
  for (int i = 0; i < 8; ++i) {
    int lm = wm * 16 + i + rsel;
    int tr = rows[lm];
    if (tr < 0) continue;
    float pv = topP[tr];
    size_t rb = (size_t)tr * D + nBase + wn * 64 + l15;
#pragma unroll
    for (int j = 0; j < 4; ++j) Out[rb + j * 16] = pv * (acc[j][i] + bv[j]);
  }
}

// ---------------- Residual + mean over S (two-stage, deterministic) ---------
__global__ __launch_bounds__(512) void mean_partial_kernel(
    const float* __restrict__ H, const float* __restrict__ O3,
    float* __restrict__ partial) {
  const int f = threadIdx.x;            // 0..511
  const int b = blockIdx.y;             // 0..7
  const int c = blockIdx.x;             // 0..31 (chunks of 64 sequence steps)
  size_t base = ((size_t)b * 2048 + (size_t)c * 64) * D + f;
  float acc = 0.f;
  for (int s = 0; s < 64; ++s) {
    size_t p = base + (size_t)s * D;
    acc += H[p] + O3[p];
  }
  partial[((size_t)b * 32 + c) * D + f] = acc;
}

__global__ __launch_bounds__(512) void sentence_kernel(
    const float* __restrict__ partial, float* __restrict__ sentence) {
  const int f = threadIdx.x;
  const int b = blockIdx.x;
  float acc = 0.f;
  for (int c = 0; c < 32; ++c) acc += partial[((size_t)b * 32 + c) * D + f];
  sentence[b * D + f] = acc * (1.0f / 2048.0f);
}

// ---------------- log-softmax + NLL loss (one block, 8 waves) ---------------
__global__ __launch_bounds__(256) void loss_kernel(
    const float* __restrict__ sentence, const int* __restrict__ y,
    float* __restrict__ out) {
  __shared__ float wloss[8];
  const int lane = threadIdx.x & 31, wave = threadIdx.x >> 5;
  const int b = wave;
  const float* row = sentence + b * D;
  float m = -3.402823466e+38f;
  for (int i = lane; i < D; i += 32) m = fmaxf(m, row[i]);
  for (int off = 16; off > 0; off >>= 1) m = fmaxf(m, __shfl_xor(m, off, 32));
  float s = 0.f;
  for (int i = lane; i < D; i += 32) s += expf(row[i] - m);
  for (int off = 16; off > 0; off >>= 1) s += __shfl_xor(s, off, 32);
  if (lane == 0) wloss[b] = -(row[y[b]] - (m + logf(s)));
  __syncthreads();
  if (threadIdx.x == 0) {
    float L = 0.f;
    for (int i = 0; i < 8; ++i) L += wloss[i];
    out[0] = L * (1.0f / 8.0f);
  }
}

// ---------------------------------------------------------------------------
extern "C" void kernel_launch(void* const* d_in, const int* in_sizes, int n_in,
                              void* d_out, int out_size, void* d_ws, size_t ws_size,
                              hipStream_t stream) {
  (void)in_sizes; (void)n_in; (void)out_size; (void)ws_size;
  const float* x   = (const float*)d_in[0];
  const float* W1  = (const float*)d_in[1];
  const float* b1  = (const float*)d_in[2];
  const float* gw2 = (const float*)d_in[3];
  const float* We2 = (const float*)d_in[4];
  const float* be2 = (const float*)d_in[5];
  const float* gw3 = (const float*)d_in[6];
  const float* We3 = (const float*)d_in[7];
  const float* be3 = (const float*)d_in[8];
  const int*   y   = (const int*)d_in[9];
  float* loss = (float*)d_out;

  // Workspace layout
  char* ws = (char*)d_ws;
  const size_t ACT = (size_t)T_TOK * D * sizeof(float);   // 33.5 MB
  float* h    = (float*)(ws);
  float* out2 = (float*)(ws + ACT);
  float* out3 = (float*)(ws + 2 * ACT);
  char*  p    = ws + 3 * ACT;
  int*   idx2 = (int*)(p);                 p += T_TOK * 4;
  float* p2   = (float*)(p);               p += T_TOK * 4;
  int*   idx3 = (int*)(p);                 p += T_TOK * 4;
  float* p3   = (float*)(p);               p += T_TOK * 4;
  int*   perm2= (int*)(p);                 p += T_TOK * 4;
  int*   perm3= (int*)(p);                 p += T_TOK * 4;
  float* partial  = (float*)(p);           p += 8 * 32 * D * 4;
  float* sentence = (float*)(p);           p += 8 * D * 4;
  int*   ctrl = (int*)(p);  // [0:8) counts2 [8:16) cursor2 [16:24) bases2
                            // [24:32) counts3 [32:40) cursor3 [40:48) bases3
  int* counts2 = ctrl, *cursor2 = ctrl + 8,  *bases2 = ctrl + 16;
  int* counts3 = ctrl + 24, *cursor3 = ctrl + 32, *bases3 = ctrl + 40;

  hipMemsetAsync(ctrl, 0, 48 * sizeof(int), stream);

  dim3 blk(256);
  dim3 gemmGrid(D / BN, T_TOK / BM);          // (4, 256)
  dim3 moeGrid(D / BN, T_TOK / BM, NEXP);     // (4, 256, 8)

  // 1) h = x @ W1^T + b1
  linear1_kernel<<<gemmGrid, blk, 0, stream>>>(x, W1, b1, h);

  // 2) MoE layer 2
  gate_kernel<<<T_TOK / 8, blk, 0, stream>>>(h, gw2, idx2, p2);
  histo_kernel<<<T_TOK / 256, blk, 0, stream>>>(idx2, counts2);
  scan_kernel<<<1, 32, 0, stream>>>(counts2, bases2, cursor2);
  scatter_kernel<<<T_TOK / 256, blk, 0, stream>>>(idx2, cursor2, perm2);
  moe_kernel<<<moeGrid, blk, 0, stream>>>(h, We2, be2, perm2, bases2, counts2, p2, out2);

  // 3) MoE layer 3
  gate_kernel<<<T_TOK / 8, blk, 0, stream>>>(out2, gw3, idx3, p3);
  histo_kernel<<<T_TOK / 256, blk, 0, stream>>>(idx3, counts3);
  scan_kernel<<<1, 32, 0, stream>>>(counts3, bases3, cursor3);
  scatter_kernel<<<T_TOK / 256, blk, 0, stream>>>(idx3, cursor3, perm3);
  moe_kernel<<<moeGrid, blk, 0, stream>>>(out2, We3, be3, perm3, bases3, counts3, p3, out3);

  // 4) hidden = h + out3 ; mean over S ; log-softmax ; NLL
  mean_partial_kernel<<<dim3(32, 8), dim3(512), 0, stream>>>(h, out3, partial);
  sentence_kernel<<<8, dim3(512), 0, stream>>>(partial, sentence);
  loss_kernel<<<1, blk, 0, stream>>>(sentence, y, loss);
}